// DivergenceScore_27462020891103
// MI455X (gfx1250) — compile-verified
//
#include <hip/hip_runtime.h>
#include <cstdint>

// Problem constants (match reference)
constexpr int N = 131072;
constexpr int D = 256;
constexpr int C = 1000;
constexpr float EPS = 1e-6f;

// Tiling
constexpr int NCHUNK = 4;               // D split into 4 chunks of 64 cols
constexpr int DCH    = D / NCHUNK;      // 64
constexpr int WGS    = 32;              // row-groups (workgroups) per chunk
constexpr int ROWS_PER_WG = N / WGS;    // 4096
constexpr int BLK    = 256;             // threads per block (8 waves of 32)
constexpr int NWAVE  = BLK / 32;        // 8
constexpr int CD     = C * DCH;         // 64000 elements per partial

// Workspace layout (bytes)
constexpr size_t P_BYTES    = (size_t)NCHUNK * WGS * CD * 4;  // 32,768,000
constexpr size_t CNTP_OFF   = P_BYTES;                        // WGS*C u32
constexpr size_t CNTT_OFF   = CNTP_OFF + (size_t)WGS * C * 4; // C u32
constexpr size_t NPRES_OFF  = CNTT_OFF + (size_t)C * 4;       // 1 u32
constexpr size_t BSUM_OFF   = NPRES_OFF + 16;                 // per-block sums (pass2)

// Dynamic LDS for pass1: sums (C*DCH f32) + staged labels + counts
constexpr size_t SMEM_SUMS   = (size_t)CD * 4;            // 256,000
constexpr size_t SMEM_LBL    = (size_t)ROWS_PER_WG * 4;   // 16,384
constexpr size_t SMEM_CNT    = (size_t)C * 4;             // 4,000
constexpr size_t SMEM1_BYTES = SMEM_SUMS + SMEM_LBL + SMEM_CNT; // 276,384 <= 320KB

typedef __attribute__((ext_vector_type(2))) float v2f;
typedef __attribute__((ext_vector_type(8))) float v8f;
typedef __attribute__((ext_vector_type(4))) int   v4i;

// Address-space-qualified vector pointers for the async global->LDS builtin
typedef __attribute__((address_space(1))) v4i* gptr_v4i;
typedef __attribute__((address_space(3))) v4i* lptr_v4i;

#if defined(__has_builtin)
#if __has_builtin(__builtin_amdgcn_global_load_async_to_lds_b128)
#define HAVE_ASYNC_LDS 1
#endif
#endif

// ---------------------------------------------------------------------------
// Deterministic 256-thread block reduction using v_wmma_f32_16x16x4_f32.
// Wave 0: lane l holds s_l; A(16x4): VGPR0=s_l (K=0 lanes0-15, K=2 lanes16-31),
// VGPR1=0; B = all-ones => D[m,n] = s_m + s_{m+16}. Sum of the 8 C/D VGPRs at
// lane0 (M=0..7) + lane16 (M=8..15) = total of all 32 s_l.
// ---------------------------------------------------------------------------
__device__ __forceinline__ float block_reduce_256(float v) {
    __shared__ float red[BLK];
    red[threadIdx.x] = v;
    __syncthreads();
    float total = 0.0f;
    if (threadIdx.x < 32) {
        float s = 0.0f;
#pragma unroll
        for (int i = 0; i < 8; ++i) s += red[threadIdx.x + 32 * i];
        v2f A; A[0] = s;    A[1] = 0.0f;
        v2f B; B[0] = 1.0f; B[1] = 1.0f;
        v8f Cz = {0.f, 0.f, 0.f, 0.f, 0.f, 0.f, 0.f, 0.f};
        v8f Dm = __builtin_amdgcn_wmma_f32_16x16x4_f32(
            false, A, false, B, (short)0, Cz, false, false);
        float t = 0.0f;
#pragma unroll
        for (int i = 0; i < 8; ++i) t += Dm[i];
        total = __shfl(t, 0, 32) + __shfl(t, 16, 32);
    }
    return total;  // valid on thread 0
}

// ---------------------------------------------------------------------------
// Pass 0: zero the n_present accumulator.
// ---------------------------------------------------------------------------
__global__ void k_init(unsigned* nPresent) {
    if (threadIdx.x == 0 && blockIdx.x == 0) *nPresent = 0u;
}

// ---------------------------------------------------------------------------
// Pass 1: streaming segment-sum into LDS (ds_add_f32), flush to partials.
// grid = NCHUNK*WGS blocks of 256 threads. Each block owns a 250KB LDS
// accumulator for its 64-column chunk and 4096 rows. Each wave processes one
// full row per iteration: lane l loads float2 {2l, 2l+1} (b64, coalesced) and
// does two ds_add_f32 hitting the 32 even then 32 odd banks (conflict-free);
// the per-row label is wave-uniform (LDS broadcast).
// ---------------------------------------------------------------------------
__global__ void __launch_bounds__(BLK)
k_segsum(const float* __restrict__ feats, const int* __restrict__ lbl,
         float* __restrict__ P, unsigned* __restrict__ CntP) {
    extern __shared__ char smem[];
    float*    sums = (float*)smem;
    int*      slbl = (int*)(smem + SMEM_SUMS);
    unsigned* scnt = (unsigned*)(smem + SMEM_SUMS + SMEM_LBL);

    const int chunk = blockIdx.x / WGS;
    const int wg    = blockIdx.x % WGS;
    const int t     = threadIdx.x;
    const int rowStart = wg * ROWS_PER_WG;

    // Zero LDS accumulators (b128).
    float4* sums4 = (float4*)sums;
    for (int i = t; i < CD / 4; i += BLK)
        sums4[i] = make_float4(0.f, 0.f, 0.f, 0.f);
    if (chunk == 0) for (int i = t; i < C; i += BLK) scnt[i] = 0u;

    // Stage this block's 4096 labels (16KB) into LDS via async b128 copies.
#if HAVE_ASYNC_LDS
    {
        const char* gsrc = (const char*)(lbl + rowStart);
        char*       ldst = (char*)slbl;
        for (int i = t * 16; i < ROWS_PER_WG * 4; i += BLK * 16) {
            __builtin_amdgcn_global_load_async_to_lds_b128(
                (gptr_v4i)(void*)(gsrc + i),
                (lptr_v4i)(void*)(ldst + i),
                0, 0);
        }
#if __has_builtin(__builtin_amdgcn_s_wait_asynccnt)
        __builtin_amdgcn_s_wait_asynccnt(0);
#else
        asm volatile("s_wait_asynccnt 0" ::: "memory");
#endif
    }
#else
    for (int i = t; i < ROWS_PER_WG; i += BLK) slbl[i] = lbl[rowStart + i];
#endif
    __syncthreads();

    // Main streaming loop: one row per wave per iteration.
    const int wave = t >> 5;        // 0..7
    const int lane = t & 31;
    const float2* fbase =
        (const float2*)(feats + (size_t)rowStart * D + (size_t)chunk * DCH) + lane;
    const bool countLane = (chunk == 0) && (lane == 0);

#pragma unroll 4
    for (int r = wave; r < ROWS_PER_WG; r += NWAVE) {
        const int    lab = slbl[r];                    // wave-uniform broadcast
        const float2 v   = fbase[(size_t)r * (D / 2)]; // global_load_b64
        float* dst = &sums[lab * DCH + 2 * lane];
        atomicAdd(dst,     v.x);                       // ds_add_f32 (even banks)
        atomicAdd(dst + 1, v.y);                       // ds_add_f32 (odd banks)
        if (countLane) atomicAdd(&scnt[lab], 1u);      // ds_add_u32
    }
    __syncthreads();

    // Non-atomic b128 flush of private accumulator to L2-resident partials.
    float4* Pb4 = (float4*)(P + ((size_t)chunk * WGS + wg) * CD);
    for (int i = t; i < CD / 4; i += BLK) Pb4[i] = sums4[i];
    if (chunk == 0) {
        unsigned* Cb = CntP + (size_t)wg * C;
        for (int i = t; i < C; i += BLK) Cb[i] = scnt[i];
    }
}

// ---------------------------------------------------------------------------
// Pass 1.5: total per-class counts + exact integer n_present.
// ---------------------------------------------------------------------------
__global__ void k_counts(const unsigned* __restrict__ CntP,
                         unsigned* __restrict__ cntTotal,
                         unsigned* __restrict__ nPresent) {
    const int c = blockIdx.x * BLK + threadIdx.x;
    if (c < C) {
        unsigned s = 0;
#pragma unroll
        for (int w = 0; w < WGS; ++w) s += CntP[(size_t)w * C + c];
        cntTotal[c] = s;
        if (s > 0) atomicAdd(nPresent, 1u);  // exact, order-independent
    }
}

// ---------------------------------------------------------------------------
// Pass 2: reduce 32 partials per (c,d), form per-element GSS term, WMMA
// block-reduce, write one deterministic partial sum per block.
// grid = NCHUNK*C*DCH / 256 = 1000 blocks.
// ---------------------------------------------------------------------------
__global__ void __launch_bounds__(BLK)
k_loss(const float* __restrict__ P, const unsigned* __restrict__ cntTotal,
       const float* __restrict__ proto, const float* __restrict__ cov,
       float* __restrict__ blockSums) {
    const int j     = blockIdx.x * BLK + threadIdx.x;  // 0..255999
    const int chunk = j / CD;
    const int rem   = j % CD;
    const int c     = rem / DCH;
    const int col   = rem % DCH;

    const float* Pb = P + (size_t)chunk * WGS * CD + rem;
    float s = 0.0f;
#pragma unroll
    for (int w = 0; w < WGS; ++w) s += Pb[(size_t)w * CD];

    const unsigned cnt  = cntTotal[c];
    const float    mean = s / fmaxf((float)cnt, 1.0f);
    const int      d    = chunk * DCH + col;
    const float    diff = mean - proto[(size_t)c * D + d];
    const float    pe   = (cnt > 0u)
                        ? (diff * diff) / (cov[(size_t)c * D + d] + EPS)
                        : 0.0f;

    const float tot = block_reduce_256(pe);
    if (threadIdx.x == 0) blockSums[blockIdx.x] = tot;
}

// ---------------------------------------------------------------------------
// Pass 3: final deterministic reduction of 1000 block sums -> scalar loss.
// ---------------------------------------------------------------------------
__global__ void __launch_bounds__(BLK)
k_final(const float* __restrict__ blockSums, const unsigned* __restrict__ nPresent,
        float* __restrict__ out) {
    float s = 0.0f;
    for (int i = threadIdx.x; i < NCHUNK * CD / BLK; i += BLK) s += blockSums[i];
    const float tot = block_reduce_256(s);
    if (threadIdx.x == 0) out[0] = tot / ((float)(*nPresent) * (float)D);
}

// ---------------------------------------------------------------------------
extern "C" void kernel_launch(void* const* d_in, const int* in_sizes, int n_in,
                              void* d_out, int out_size, void* d_ws, size_t ws_size,
                              hipStream_t stream) {
    (void)in_sizes; (void)n_in; (void)out_size; (void)ws_size;

    const float* feats = (const float*)d_in[0];
    const int*   lbl   = (const int*)d_in[1];
    const float* proto = (const float*)d_in[2];
    const float* cov   = (const float*)d_in[3];
    float*       out   = (float*)d_out;

    char*     ws       = (char*)d_ws;
    float*    P        = (float*)(ws);
    unsigned* CntP     = (unsigned*)(ws + CNTP_OFF);
    unsigned* cntTotal = (unsigned*)(ws + CNTT_OFF);
    unsigned* nPresent = (unsigned*)(ws + NPRES_OFF);
    float*    bSums    = (float*)(ws + BSUM_OFF);

    k_init<<<1, 1, 0, stream>>>(nPresent);

    k_segsum<<<NCHUNK * WGS, BLK, (unsigned)SMEM1_BYTES, stream>>>(feats, lbl, P, CntP);

    k_counts<<<(C + BLK - 1) / BLK, BLK, 0, stream>>>(CntP, cntTotal, nPresent);

    k_loss<<<(NCHUNK * CD) / BLK, BLK, 0, stream>>>(P, cntTotal, proto, cov, bSums);

    k_final<<<1, BLK, 0, stream>>>(bSums, nPresent, out);
}